// DualSystemsModel_7670811590769
// MI455X (gfx1250) — compile-verified
//
#include <hip/hip_runtime.h>
#include <math.h>

#define IGN_VAL (-100)
#define PPB 32               // participants per block = one wave32 (barrier-free)
#define TT  16               // trials per LDS tile (24 async b128 issues/tile -> 2 tiles in flight <= 63 ASYNCcnt)
#define ROWD (TT * 2 + 4)    // dwords per LDS row: 32 data + 4 pad (144B, 16B-aligned stride)
#define BUFD (PPB * ROWD)    // dwords per buffer

// ---- prologue kernel 1: last valid (clipped) first-stage choice per participant
__global__ void last_valid_kernel(const int* __restrict__ choice,
                                  int* __restrict__ lastA1, int Pn, int T) {
  int p = blockIdx.x * blockDim.x + threadIdx.x;
  if (p >= Pn) return;
  int lv = -1;
  for (int t = T - 1; t >= 0; --t) {
    int a1 = choice[(((size_t)p * T) + t) * 2 + 0];
    if (a1 != IGN_VAL) { lv = a1 < 0 ? 0 : (a1 > 1 ? 1 : a1); break; }
  }
  lastA1[p] = lv;
}

// ---- prologue kernel 2: serial carry over participants -> rep0[p] one-hot
__global__ void carry_kernel(const int* __restrict__ lastA1,
                             float* __restrict__ rep0, int Pn) {
  if (blockIdx.x != 0 || threadIdx.x != 0) return;
  int carry = -1;
  for (int p = 0; p < Pn; ++p) {
    rep0[2 * p + 0] = (carry == 0) ? 1.0f : 0.0f;
    rep0[2 * p + 1] = (carry == 1) ? 1.0f : 0.0f;
    int lv = lastA1[p];
    if (lv >= 0) carry = lv;
  }
}

// ---- main kernel: one wave32 per block, one thread per participant.
// Trials stream through a double-buffered LDS tile filled by CDNA5 async
// global->LDS b128 copies; s_wait_asynccnt pipelines copy vs compute.
__global__ __launch_bounds__(PPB) void dualsys_kernel(
    const int* __restrict__ choice, const float* __restrict__ state,
    const float* __restrict__ reward,
    const float* __restrict__ g_tau, const float* __restrict__ g_alpha,
    const float* __restrict__ g_lambd, const float* __restrict__ g_stick,
    const float* __restrict__ g_beta,
    const float* __restrict__ rep0,
    float* __restrict__ out, int Pn, int T) {
  __shared__ int   sC[2 * BUFD];
  __shared__ float sS[2 * BUFD];
  __shared__ float sR[2 * BUFD];

  const int tid = threadIdx.x;
  const int p0  = blockIdx.x * PPB;
  const int p   = p0 + tid;
  const bool active = (p < Pn);

  const float tau_s   = 1.0f / (1.0f + expf(-g_tau[0]));
  const float alpha_s = 1.0f / (1.0f + expf(-g_alpha[0]));
  const float lambd_s = 1.0f / (1.0f + expf(-g_lambd[0]));
  const float stick   = tanhf(g_stick[0]);
  const float beta    = g_beta[0];

  float q00 = 0.f, q01 = 0.f, q10 = 0.f, q11 = 0.f, q20 = 0.f, q21 = 0.f;
  float r0 = 0.f, r1 = 0.f;
  if (active) { r0 = rep0[2 * p + 0]; r1 = rep0[2 * p + 1]; }

  // Low 32 bits of a generic pointer to __shared__ = LDS byte offset.
  const unsigned ldsC = (unsigned)(size_t)(void*)sC;
  const unsigned ldsS = (unsigned)(size_t)(void*)sS;
  const unsigned ldsR = (unsigned)(size_t)(void*)sR;

  const size_t rowElems = (size_t)T * 2;
  const int nTiles = T / TT;  // T is a multiple of TT (T=512)

  // Issue one full tile (24 async b128 ops per thread: 3 arrays x 8 chunks).
  // Lane->chunk map: consecutive lanes take consecutive 16B chunks, covering
  // contiguous 128B runs of each participant row in global memory.
  auto issue_tile = [&](int t0, int buf) {
    if (t0 > T - TT) t0 = T - TT;  // clamp: always a full, in-bounds tile
    if (t0 < 0) t0 = 0;
#pragma unroll
    for (int k = 0; k < 8; ++k) {
      int c   = tid + k * PPB;   // 0..255
      int row = c >> 3;
      int qk  = c & 7;
      int gRow = p0 + row;
      if (gRow >= Pn) gRow = p0;
      size_t   gOff = ((size_t)gRow * rowElems + (size_t)t0 * 2) * 4 + (size_t)qk * 16;
      unsigned lOff = (unsigned)(buf * (BUFD * 4) + row * (ROWD * 4) + qk * 16);
      const char* gc = (const char*)choice + gOff;
      const char* gs = (const char*)state  + gOff;
      const char* gr = (const char*)reward + gOff;
      unsigned dc  = ldsC + lOff;
      unsigned dsd = ldsS + lOff;
      unsigned dr  = ldsR + lOff;
      asm volatile("global_load_async_to_lds_b128 %0, %1, off" ::"v"(dc),  "v"(gc) : "memory");
      asm volatile("global_load_async_to_lds_b128 %0, %1, off" ::"v"(dsd), "v"(gs) : "memory");
      asm volatile("global_load_async_to_lds_b128 %0, %1, off" ::"v"(dr),  "v"(gr) : "memory");
    }
  };

  issue_tile(0, 0);  // prologue fill of buffer 0

  for (int ti = 0; ti < nTiles; ++ti) {
    // Prefetch next tile into the other buffer (dummy re-issue on last iter).
    int tn = (ti + 1 < nTiles) ? (ti + 1) : ti;
    issue_tile(tn * TT, (ti + 1) & 1);

    // 48 outstanding; wait down to 24 -> tile ti (issued first, in-order) done.
    asm volatile("s_wait_asynccnt 0x18" ::: "memory");

    const int t0   = ti * TT;
    const int bufO = (ti & 1) * BUFD;
    if (active) {
      const int base = bufO + tid * ROWD;
#pragma unroll 4
      for (int tt = 0; tt < TT; ++tt) {
        int   a1 = sC[base + 2 * tt + 0];
        int   a2 = sC[base + 2 * tt + 1];
        float sf = sS[base + 2 * tt + 1];
        float rr = sR[base + 2 * tt + 1];

        bool v1 = (a1 != IGN_VAL);
        bool v2 = (a2 != IGN_VAL);
        int a1c = a1 < 0 ? 0 : (a1 > 1 ? 1 : a1);
        int a2c = a2 < 0 ? 0 : (a2 > 1 ? 1 : a2);
        int sv  = (int)sf;
        int sc  = sv < 0 ? 0 : (sv > 1 ? 1 : sv);
        bool sIs1 = (sc == 0);  // srow == 1

        // model-based values through fixed transition matrix [[.7,.3],[.3,.7]]
        float m0 = fmaxf(q10, q11);
        float m1 = fmaxf(q20, q21);
        float qmb0 = 0.7f * m0 + 0.3f * m1;
        float qmb1 = 0.3f * m0 + 0.7f * m1;
        float qnet0 = tau_s * qmb0 + (1.0f - tau_s) * q00;
        float qnet1 = tau_s * qmb1 + (1.0f - tau_s) * q01;

        float qs0 = sIs1 ? q10 : q20;
        float qs1 = sIs1 ? q11 : q21;

        float l00 = v1 ? (qnet0 + r0 * stick) : 0.0f;
        float l01 = v1 ? (qnet1 + r1 * stick) : 0.0f;
        float l10 = v1 ? qs0 : 0.0f;
        float l11 = v1 ? qs1 : 0.0f;

        float4 o;
        o.x = l00 * beta; o.y = l01 * beta; o.z = l10 * beta; o.w = l11 * beta;
        *reinterpret_cast<float4*>(&out[(((size_t)p * T) + (t0 + tt)) * 4]) = o;

        // updates (match reference's sequential .at[].add order)
        float q0a = a1c ? q01 : q00;
        float qsa = a2c ? qs1 : qs0;
        float d1 = qsa - q0a;
        float d2 = rr - qsa;
        bool upd = v1 && v2;
        float addA = alpha_s * d1;
        float addB = lambd_s * (alpha_s * d2);
        float addS = alpha_s * d2;

        q00 = (upd && a1c == 0) ? ((q00 + addA) + addB) : q00;
        q01 = (upd && a1c == 1) ? ((q01 + addA) + addB) : q01;
        q10 = (upd &&  sIs1 && a2c == 0) ? (q10 + addS) : q10;
        q11 = (upd &&  sIs1 && a2c == 1) ? (q11 + addS) : q11;
        q20 = (upd && !sIs1 && a2c == 0) ? (q20 + addS) : q20;
        q21 = (upd && !sIs1 && a2c == 1) ? (q21 + addS) : q21;

        r0 = v1 ? (a1c == 0 ? 1.0f : 0.0f) : r0;
        r1 = v1 ? (a1c == 1 ? 1.0f : 0.0f) : r1;
      }
    }
  }
  // leftover dummy-prefetch async ops are drained by S_ENDPGM's implicit wait.
}

extern "C" void kernel_launch(void* const* d_in, const int* in_sizes, int n_in,
                              void* d_out, int out_size, void* d_ws, size_t ws_size,
                              hipStream_t stream) {
  const int*   choice = (const int*)d_in[0];
  const float* state  = (const float*)d_in[1];
  const float* reward = (const float*)d_in[2];
  const float* tau    = (const float*)d_in[3];
  const float* alpha  = (const float*)d_in[4];
  const float* lambd  = (const float*)d_in[5];
  const float* stick  = (const float*)d_in[6];
  const float* beta   = (const float*)d_in[7];
  float* out = (float*)d_out;

  const int Pn = 4096;
  const int T  = in_sizes[0] / (Pn * 2);

  int*   lastA1 = (int*)d_ws;
  float* rep0   = (float*)((char*)d_ws + (((size_t)Pn * sizeof(int) + 255) & ~(size_t)255));

  last_valid_kernel<<<(Pn + 255) / 256, 256, 0, stream>>>(choice, lastA1, Pn, T);
  carry_kernel<<<1, 1, 0, stream>>>(lastA1, rep0, Pn);
  dualsys_kernel<<<(Pn + PPB - 1) / PPB, PPB, 0, stream>>>(
      choice, state, reward, tau, alpha, lambd, stick, beta, rep0, out, Pn, T);
}